// MIE_21423296872467
// MI455X (gfx1250) — compile-verified
//
#include <hip/hip_runtime.h>
#include <hip/hip_bf16.h>

// MIE loss for N=16,777,216 samples, NUM_PER_BIN=50, NB=N/50=335,544 bins.
//
// Core identity (tie-exact vs the reference's searchsorted-left semantics):
//     bin(v) = min(strict_rank(v) / 50, NB-1)
// so no sort is needed: bucket by floor(v * 2^20) (monotone in value),
// exclusive-scan bucket counts for base ranks, then resolve within tiny
// buckets (mean 16 elements) with one wave32 per bucket using async
// global->LDS copies (CDNA5 ASYNCcnt path). All cross-block accumulation is
// integer atomics -> deterministic output. The final 1024-partial fp32
// reduction runs through V_WMMA_F32_16X16X4_F32 (ones-matrix row-sum),
// which is exact fp32 and has a fixed hardware accumulation order.

#define NPB           50
#define LOG_BUCKETS   20
#define BUCKETS       (1 << LOG_BUCKETS)   // 1,048,576
#define CAP           64                   // per-wave LDS record capacity
#define WAVES_PER_BLK 8

typedef __attribute__((ext_vector_type(2))) float v2f;
typedef __attribute__((ext_vector_type(8))) float v8f;

// ---------------------------------------------------------------- helpers

__device__ __forceinline__ unsigned key_of(float f) {
  unsigned u = __float_as_uint(f);
  // order-preserving fp32 -> u32 (handles sign bit robustly)
  return u ^ ((unsigned)((int)u >> 31) | 0x80000000u);
}

__device__ __forceinline__ int bucket_of(float f) {
  float x = f * (float)BUCKETS;      // monotone for f >= 0
  int b = (int)x;
  b = (b < 0) ? 0 : b;
  b = (b >= BUCKETS) ? (BUCKETS - 1) : b;
  return b;
}

__device__ __forceinline__ float bin_entropy(float p) {
  float pl = (p > 0.0f) ? p * log2f(fmaxf(p, 1e-12f)) : 0.0f;
  float q  = 1.0f - p;
  float ql = (p < 1.0f) ? q * log2f(fmaxf(q, 1e-12f)) : 0.0f;
  return -(pl + ql);
}

// ---------------------------------------------------------------- kernels

__global__ void k_zero(unsigned* p, size_t nWords) {
  size_t i      = (size_t)blockIdx.x * blockDim.x + threadIdx.x;
  size_t stride = (size_t)gridDim.x * blockDim.x;
  for (; i < nWords; i += stride) p[i] = 0u;
}

// histogram over 2^20 buckets + total accuracy sum (integer atomics: deterministic)
__global__ void k_hist(const float* __restrict__ conf, const int* __restrict__ acc,
                       unsigned* __restrict__ bucketCount, unsigned* __restrict__ totalAcc,
                       int n) {
  int i      = blockIdx.x * blockDim.x + threadIdx.x;
  int stride = gridDim.x * blockDim.x;
  unsigned accSum = 0;
  for (; i < n; i += stride) {
    atomicAdd(&bucketCount[bucket_of(conf[i])], 1u);
    accSum += (unsigned)acc[i];
  }
  for (int off = 16; off > 0; off >>= 1) accSum += __shfl_down(accSum, off, 32);
  if ((threadIdx.x & 31) == 0) atomicAdd(totalAcc, accSum);
}

// exclusive scan of BUCKETS counts: 3-kernel two-level scan (1024 x 1024)
__global__ void k_scan_partials(const unsigned* __restrict__ counts,
                                unsigned* __restrict__ blockSums) {
  __shared__ unsigned s[256];
  int b = blockIdx.x, t = threadIdx.x;
  const unsigned* p = counts + (size_t)b * 1024;
  unsigned v = p[t] + p[t + 256] + p[t + 512] + p[t + 768];
  s[t] = v; __syncthreads();
  for (int off = 128; off > 0; off >>= 1) {
    if (t < off) s[t] += s[t + off];
    __syncthreads();
  }
  if (t == 0) blockSums[b] = s[0];
}

__global__ void k_scan_block(unsigned* blockSums) {  // in-place exclusive scan of 1024
  __shared__ unsigned s[1024];
  int t = threadIdx.x;
  for (int k = 0; k < 4; ++k) s[t + 256 * k] = blockSums[t + 256 * k];
  __syncthreads();
  if (t == 0) {
    unsigned run = 0;
    for (int i = 0; i < 1024; ++i) { unsigned c = s[i]; s[i] = run; run += c; }
  }
  __syncthreads();
  for (int k = 0; k < 4; ++k) blockSums[t + 256 * k] = s[t + 256 * k];
}

__global__ void k_scan_final(const unsigned* __restrict__ counts,
                             const unsigned* __restrict__ blockOffs,
                             unsigned* __restrict__ base) {
  __shared__ unsigned sc[256];
  int b = blockIdx.x, t = threadIdx.x;
  const unsigned* p = counts + (size_t)b * 1024;
  unsigned a0 = p[4 * t], a1 = p[4 * t + 1], a2 = p[4 * t + 2], a3 = p[4 * t + 3];
  unsigned tsum = a0 + a1 + a2 + a3;
  sc[t] = tsum; __syncthreads();
  for (int off = 1; off < 256; off <<= 1) {           // inclusive Hillis-Steele
    unsigned add = (t >= off) ? sc[t - off] : 0u;
    unsigned v   = sc[t];
    __syncthreads();
    sc[t] = v + add;
    __syncthreads();
  }
  unsigned excl = ((t == 0) ? 0u : sc[t - 1]) + blockOffs[b];
  unsigned* out = base + (size_t)b * 1024;
  out[4 * t]     = excl;
  out[4 * t + 1] = excl + a0;
  out[4 * t + 2] = excl + a0 + a1;
  out[4 * t + 3] = excl + a0 + a1 + a2;
}

// scatter (key,acc) records bucket-contiguously; order within a bucket irrelevant
__global__ void k_scatter(const float* __restrict__ conf, const int* __restrict__ acc,
                          const unsigned* __restrict__ base, unsigned* __restrict__ cursor,
                          unsigned long long* __restrict__ records, int n) {
  int i      = blockIdx.x * blockDim.x + threadIdx.x;
  int stride = gridDim.x * blockDim.x;
  for (; i < n; i += stride) {
    float f = conf[i];
    int b = bucket_of(f);
    unsigned pos = base[b] + atomicAdd(&cursor[b], 1u);
    records[pos] = ((unsigned long long)key_of(f) << 32) | (unsigned long long)(acc[i] & 1);
  }
}

// one wave32 per bucket: async-copy records to LDS, strict within-bucket ranks
// via O(m^2/32) compares, then bin = min((base+rank)/50, NB-1) aggregates.
__global__ void __launch_bounds__(256) k_rank(const unsigned* __restrict__ counts,
                                              const unsigned* __restrict__ base,
                                              const unsigned long long* __restrict__ records,
                                              unsigned* __restrict__ binSize,
                                              unsigned* __restrict__ binAcc,
                                              int nb) {
  __shared__ unsigned long long lds[WAVES_PER_BLK * CAP];
  int t = threadIdx.x;
  int wave = t >> 5, lane = t & 31;
  int bucket = blockIdx.x * WAVES_PER_BLK + wave;     // grid sized exactly: wave-uniform
  unsigned m = counts[bucket];
  if (m == 0) return;                                  // wave-uniform branch
  unsigned b0 = base[bucket];
  unsigned nbm1 = (unsigned)(nb - 1);

  if (m <= CAP) {
    unsigned long long* wlds = lds + wave * CAP;
    // CDNA5 async global -> LDS copy (ASYNCcnt). IOFFSET applies to both the
    // LDS and memory addresses, so offset:256 moves lanes to records [32..63].
    unsigned ldsoff = (unsigned)(unsigned long long)(void*)&wlds[lane];
    unsigned long long gaddr =
        (unsigned long long)(const void*)(records + b0) + (unsigned long long)lane * 8ull;
    asm volatile("global_load_async_to_lds_b64 %0, %1, off\n\t"
                 "global_load_async_to_lds_b64 %0, %1, off offset:256"
                 :: "v"(ldsoff), "v"(gaddr) : "memory");
    asm volatile("s_wait_asynccnt 0" ::: "memory");

    for (unsigned e = lane; e < m; e += 32) {
      unsigned long long re = wlds[e];
      unsigned ke = (unsigned)(re >> 32);
      unsigned r = 0;
      for (unsigned j = 0; j < m; ++j)
        r += ((unsigned)(wlds[j] >> 32) < ke) ? 1u : 0u;
      unsigned bin = (b0 + r) / NPB;
      bin = (bin > nbm1) ? nbm1 : bin;
      atomicAdd(&binSize[bin], 1u);
      atomicAdd(&binAcc[bin], (unsigned)(re & 1ull));
    }
  } else {
    // rare oversized bucket: correct O(m^2) fallback straight from L2
    for (unsigned e = lane; e < m; e += 32) {
      unsigned long long re = records[b0 + e];
      unsigned ke = (unsigned)(re >> 32);
      unsigned r = 0;
      for (unsigned j = 0; j < m; ++j)
        r += ((unsigned)(records[b0 + j] >> 32) < ke) ? 1u : 0u;
      unsigned bin = (b0 + r) / NPB;
      bin = (bin > nbm1) ? nbm1 : bin;
      atomicAdd(&binSize[bin], 1u);
      atomicAdd(&binAcc[bin], (unsigned)(re & 1ull));
    }
  }
}

// deterministic entropy reduction: fixed chunks -> fixed-order block tree
__global__ void k_entropy_partial(const unsigned* __restrict__ binSize,
                                  const unsigned* __restrict__ binAcc,
                                  float* __restrict__ partial, int nb) {
  __shared__ float sh[256];
  int b = blockIdx.x, t = threadIdx.x;
  int chunk = (nb + 1023) / 1024;
  int start = b * chunk;
  int end   = start + chunk; if (end > nb) end = nb;
  float s = 0.0f;
  for (int i = start + t; i < end; i += 256) {
    unsigned sz = binSize[i];
    if (sz) {
      float p = (float)binAcc[i] / (float)sz;
      s += (float)sz * bin_entropy(p);
    }
  }
  sh[t] = s; __syncthreads();
  for (int off = 128; off > 0; off >>= 1) {
    if (t < off) sh[t] += sh[t + off];
    __syncthreads();
  }
  if (t == 0) partial[b] = sh[0];
}

// Final reduction of 1024 fp32 partials via V_WMMA_F32_16X16X4_F32.
// A(16x4) holds 64 slot-sums (each slot = fixed-order sum of 16 partials);
// B = ones(4x16)  =>  D[i][j] = rowsum(A_i).  Per the C/D layout, lanes 0-15
// hold rows 0-7 in their 8 VGPRs, lanes 16-31 hold rows 8-15, so
// total = (sum of d[0..7] on lane 0) + (same on lane 16).  Exact fp32,
// fixed accumulation order -> deterministic. Launched as one full wave32
// (EXEC all ones, as WMMA requires).
__global__ void __launch_bounds__(32) k_final(const float* __restrict__ partial,
                                              const unsigned* __restrict__ totalAcc,
                                              float* __restrict__ out, int n) {
  int lane = threadIdx.x;                 // 0..31
  int M  = lane & 15;                     // A-matrix row for this lane
  int k0 = (lane < 16) ? 0 : 2;           // this lane holds K=k0 and K=k0+1

  // slot s (0..63) = M*4 + K;  slot value = sum of partial[s*16 .. s*16+15]
  float s0 = 0.0f, s1 = 0.0f;
  int baseIdx0 = (M * 4 + k0) * 16;
  int baseIdx1 = (M * 4 + k0 + 1) * 16;
  for (int i = 0; i < 16; ++i) {
    s0 += partial[baseIdx0 + i];
    s1 += partial[baseIdx1 + i];
  }

  v2f a; a.x = s0; a.y = s1;
  v2f b; b.x = 1.0f; b.y = 1.0f;          // ones matrix: layout-independent
  v8f c = {};
  v8f d = __builtin_amdgcn_wmma_f32_16x16x4_f32(
      /*neg_a=*/false, a, /*neg_b=*/false, b,
      /*c_mod=*/(short)0, c, /*reuse_a=*/false, /*reuse_b=*/false);

  float s = d[0] + d[1] + d[2] + d[3] + d[4] + d[5] + d[6] + d[7];
  float total = __shfl(s, 0, 32) + __shfl(s, 16, 32);

  if (lane == 0) {
    float tot = (float)(*totalAcc) / (float)n;
    out[0] = bin_entropy(tot) - total / (float)n;
  }
}

// ---------------------------------------------------------------- launcher

extern "C" void kernel_launch(void* const* d_in, const int* in_sizes, int n_in,
                              void* d_out, int out_size, void* d_ws, size_t ws_size,
                              hipStream_t stream) {
  const float* conf = (const float*)d_in[0];
  const int*   acc  = (const int*)d_in[1];
  const int n  = in_sizes[0];
  const int nb = n / NPB;

  char* ws = (char*)d_ws;
  size_t o = 0;
  auto alloc = [&](size_t bytes) -> void* {
    void* p = ws + o;
    o += (bytes + 255) & ~(size_t)255;
    return p;
  };

  // zeroed region (counters only) -----------------------------------------
  unsigned* bucketCount = (unsigned*)alloc((size_t)BUCKETS * 4);
  unsigned* cursor      = (unsigned*)alloc((size_t)BUCKETS * 4);
  unsigned* binSize     = (unsigned*)alloc((size_t)nb * 4);
  unsigned* binAcc      = (unsigned*)alloc((size_t)nb * 4);
  unsigned* totalAcc    = (unsigned*)alloc(256);
  const size_t zeroWords = o / 4;
  // fully-overwritten region ----------------------------------------------
  unsigned*           blockSums = (unsigned*)alloc(1024 * 4);
  float*              partial   = (float*)alloc(1024 * 4);
  unsigned*           base      = (unsigned*)alloc((size_t)BUCKETS * 4);
  unsigned long long* records   = (unsigned long long*)alloc(((size_t)n + CAP) * 8);
  (void)ws_size; (void)n_in; (void)out_size;

  k_zero<<<2048, 256, 0, stream>>>((unsigned*)ws, zeroWords);
  k_hist<<<4096, 256, 0, stream>>>(conf, acc, bucketCount, totalAcc, n);
  k_scan_partials<<<1024, 256, 0, stream>>>(bucketCount, blockSums);
  k_scan_block<<<1, 256, 0, stream>>>(blockSums);
  k_scan_final<<<1024, 256, 0, stream>>>(bucketCount, blockSums, base);
  k_scatter<<<4096, 256, 0, stream>>>(conf, acc, base, cursor, records, n);
  k_rank<<<BUCKETS / WAVES_PER_BLK, 256, 0, stream>>>(bucketCount, base, records,
                                                      binSize, binAcc, nb);
  k_entropy_partial<<<1024, 256, 0, stream>>>(binSize, binAcc, partial, nb);
  k_final<<<1, 32, 0, stream>>>(partial, totalAcc, (float*)d_out, n);
}